// Encoder_29661044146233
// MI455X (gfx1250) — compile-verified
//
#include <hip/hip_runtime.h>

typedef __attribute__((ext_vector_type(2))) float v2f;
typedef __attribute__((ext_vector_type(8))) float v8f;
typedef __attribute__((ext_vector_type(4))) unsigned int v4u;
typedef __attribute__((ext_vector_type(8))) int v8i_t;
typedef __attribute__((ext_vector_type(4))) int v4i_t;

#define NJ 22
#define CIN 88
#define KFLAT 352     // CIN * 4 taps
#define BCONV 192     // 6 waves of 32
#define NWAVE 6
#define NROWS_W 96    // 88 rounded up to 6 WMMA M-tiles
#define WS 354        // weight LDS row stride
#define XS 89         // packed x-patch LDS row stride (TDM tile_w, OOB col zero-fills)
#define XIS 67        // conv1 input-part LDS row stride (66 + OOB pad col)
#define XOS 23        // conv1 offset-part LDS row stride (22 + OOB pad col)
#define XBUFMAX 3060  // max(34*XS=3026, 34*XIS+34*XOS=3060)
#define YS 20         // output-transpose LDS row stride
#define GBY 4

#if defined(__gfx1250__) && __has_builtin(__builtin_amdgcn_tensor_load_to_lds) && \
    __has_builtin(__builtin_amdgcn_s_wait_tensorcnt)
#define HAVE_TDM 1
#if __has_include(<hip/amd_detail/amd_gfx1250_TDM.h>)
#define TDM_6ARG 1
#else
#define TDM_6ARG 0
#endif
#else
#define HAVE_TDM 0
#endif

__device__ __constant__ int d_PARENT[NJ] =
    {0,0,1,2,3,4,0,6,7,8,0,10,11,12,12,14,15,16,12,18,19,20};

#if HAVE_TDM
__device__ __forceinline__ unsigned lds_off(const void* p) {
  return (unsigned)(unsigned long long)(size_t)p;   // flat addr[31:0] == LDS offset
}

// 2D tensor tile -> LDS via Tensor Data Mover. data_size = 4B.
// tile_w/tile_h may exceed tensor_w/tensor_h: OOB elements zero-fill (pads LDS).
__device__ __forceinline__ void tdm_load_2d(
    const void* gptr, unsigned ldsoff,
    unsigned tensor_w, unsigned tensor_h,
    unsigned tile_w, unsigned tile_h, unsigned row_stride_elems)
{
  unsigned long long ga = (unsigned long long)gptr;
  v4u g0;
  g0.x = 1u;                                            // count=1, user descriptor
  g0.y = ldsoff;                                        // lds_addr [63:32]
  g0.z = (unsigned)ga;                                  // global_addr [95:64]
  g0.w = (unsigned)((ga >> 32) & 0x01ffffffu) | (2u << 30);  // addr[56:32] | type=2
  unsigned d0 = 2u << 16;                               // data_size = 4B
  unsigned d1 = (tensor_w & 0xffffu) << 16;             // tensor_dim0 [79:48] lo
  unsigned d2 = (tensor_w >> 16) | ((tensor_h & 0xffffu) << 16);
  unsigned d3 = (tensor_h >> 16) | ((tile_w & 0xffffu) << 16); // tile_dim0 [127:112]
  unsigned d4 = tile_h & 0xffffu;                       // tile_dim1; tile_dim2 = 0
  unsigned d5 = row_stride_elems;                       // tensor_dim0_stride lo32
  v8i_t g1 = { (int)d0, (int)d1, (int)d2, (int)d3, (int)d4, (int)d5, 0, 0 };
  v4i_t gz = { 0, 0, 0, 0 };
#if TDM_6ARG
  v8i_t g4 = { 0, 0, 0, 0, 0, 0, 0, 0 };
  __builtin_amdgcn_tensor_load_to_lds(g0, g1, gz, gz, g4, 0);
#else
  __builtin_amdgcn_tensor_load_to_lds(g0, g1, gz, gz, 0);
#endif
}

// Stage the 34-row x-patch for output tile fo0.
// SRCMODE 0: two TDMs (input rows of 66, offset rows of 22) into split regions.
// SRCMODE 1: one TDM from packed (B,F,88) buffer.
template<int SRCMODE>
__device__ __forceinline__ void issue_patch(const float* src, const float* src2,
                                            int b, int Fin, int fo0, float* xbuf)
{
  int t0 = fo0 * 2 - 1;
  int lead = (t0 < 0) ? 1 : 0;        // t = -1 row zeroed by caller
  int t_start = t0 + lead;
  unsigned th = (unsigned)(Fin - t_start);     // trailing rows OOB -> zero
  unsigned rows = (unsigned)(34 - lead);
  if (SRCMODE == 1) {
    tdm_load_2d(src + ((size_t)b * Fin + t_start) * CIN,
                lds_off(xbuf) + (unsigned)(lead * XS * 4),
                CIN, th, XS, rows, CIN);
  } else {
    tdm_load_2d(src + ((size_t)b * Fin + t_start) * (NJ * 3),
                lds_off(xbuf) + (unsigned)(lead * XIS * 4),
                NJ * 3, th, XIS, rows, NJ * 3);
    tdm_load_2d(src2 + ((size_t)b * Fin + t_start) * NJ,
                lds_off(xbuf + 34 * XIS) + (unsigned)(lead * XOS * 4),
                NJ, th, XOS, rows, NJ);
  }
}
#endif  // HAVE_TDM

// ---------------------------------------------------------------------------
// Strided conv (K=4, stride=2, pad=1) as WMMA f32 16x16x4 GEMM.
// ---------------------------------------------------------------------------
template<int SRCMODE, int DSTMODE, int FIN, int FOUT, int TPB>
__global__ __launch_bounds__(BCONV) void conv_wmma_kernel(
    const float* __restrict__ src, const float* __restrict__ src2,
    const float* __restrict__ w,   const float* __restrict__ bias,
    float* __restrict__ dst)
{
  constexpr int NT     = FOUT / 16;
  constexpr int CHUNKS = NT / TPB;      // power of two in all launches

  extern __shared__ float lds[];
  float* wlds  = lds;                     // NROWS_W * WS
  float* xlds0 = wlds + NROWS_W * WS;     // XBUFMAX (double-buffered)
  float* xlds1 = xlds0 + XBUFMAX;
  float* ylds  = xlds1 + XBUFMAX;         // NROWS_W * YS transpose tile

  const int tid  = threadIdx.x;
  const int wave = tid >> 5;
  const int lane = tid & 31;

  const int b     = (int)blockIdx.x / CHUNKS;
  const int chunk = (int)blockIdx.x % CHUNKS;
  const int tile0 = chunk * TPB;

  const int m0    = wave << 4;            // output-channel tile base
  const int nrow  = lane & 15;            // A row == B/D column index
  const int khalf = (lane >> 4) << 1;     // hi half-wave K sub-offset
  const int tl0   = 2 * nrow + khalf;

  // ---- stage weights (rows>=88, cols>=352 zero-padded) ----
#if HAVE_TDM
  if (wave == 0)
    tdm_load_2d(w, lds_off(wlds), KFLAT, CIN, WS, NROWS_W, KFLAT);
#else
  for (int co = wave; co < NROWS_W; co += NWAVE)
    for (int kk = lane; kk < WS; kk += 32)
      wlds[co * WS + kk] = (co < CIN && kk < KFLAT) ? w[co * KFLAT + kk] : 0.f;
#endif

  auto compute_store = [&](int fo0, const float* xbuf) {
    v8f acc = {};
    const float* wrow = wlds + (m0 + nrow) * WS + khalf;
    if constexpr (SRCMODE == 0) {
      const float* xin_l  = xbuf + tl0 * XIS;
      const float* xoff_l = xbuf + 34 * XIS + tl0 * XOS;
      #pragma unroll
      for (int kk = 0; kk < KFLAT; kk += 4) {
        int ci = kk >> 2, j = ci >> 2, c = ci & 3;   // channel = j*4 + c
        v2f af, bf;
        af.x = wrow[kk];
        af.y = wrow[kk + 1];
        if (c < 3) { bf.x = xin_l[j * 3 + c];  bf.y = xin_l[XIS + j * 3 + c]; }
        else       { bf.x = xoff_l[j];         bf.y = xoff_l[XOS + j]; }
        acc = __builtin_amdgcn_wmma_f32_16x16x4_f32(
            false, af, false, bf, (short)0, acc, false, false);
      }
    } else {
      const float* xcol0 = xbuf + tl0 * XS;
      #pragma unroll
      for (int kk = 0; kk < KFLAT; kk += 4) {
        int ci = kk >> 2;
        v2f af, bf;
        af.x = wrow[kk];
        af.y = wrow[kk + 1];
        bf.x = xcol0[ci];
        bf.y = xcol0[XS + ci];
        acc = __builtin_amdgcn_wmma_f32_16x16x4_f32(
            false, af, false, bf, (short)0, acc, false, false);
      }
    }
    #pragma unroll
    for (int r = 0; r < 8; ++r)
      ylds[(m0 + r + ((lane >> 4) << 3)) * YS + nrow] = acc[r];
    __syncthreads();
    if constexpr (DSTMODE == 0) {
      for (int n = wave; n < 16; n += NWAVE) {
        float* drow = dst + ((size_t)b * FOUT + (fo0 + n)) * CIN;
        for (int co = lane; co < CIN; co += 32)
          drow[co] = ylds[co * YS + n] + bias[co];
      }
    } else {
      for (int n = wave; n < 16; n += NWAVE) {
        float* drow = dst + ((size_t)b * FOUT + (fo0 + n)) * NJ * 3;
        for (int r = lane; r < 66; r += 32) {
          int jj = r / 3, c = r - jj * 3;
          int co = (jj << 2) + c;
          drow[r] = ylds[co * YS + n] + bias[co];
        }
      }
    }
  };

#if HAVE_TDM
  // Prologue: first patch (primes buffer 0; wait covers weights TDM too).
  if (wave == 0) issue_patch<SRCMODE>(src, src2, b, FIN, tile0 << 4, xlds0);
  if (tile0 == 0) {                        // zero the t = -1 lead row(s)
    if constexpr (SRCMODE == 0) {
      for (int q = tid; q < XIS; q += BCONV) xlds0[q] = 0.f;
      for (int q = tid; q < XOS; q += BCONV) xlds0[34 * XIS + q] = 0.f;
    } else {
      for (int q = tid; q < XS; q += BCONV) xlds0[q] = 0.f;
    }
  }
  if (wave == 0) __builtin_amdgcn_s_wait_tensorcnt(0);
  __syncthreads();
  for (int tt = 0; tt < TPB; ++tt) {
    int tile = tile0 + tt;
    float* xb = (tt & 1) ? xlds1 : xlds0;
    float* xn = (tt & 1) ? xlds0 : xlds1;
    if (wave == 0 && (tt + 1) < TPB)       // async prefetch of next tile
      issue_patch<SRCMODE>(src, src2, b, FIN, (tile + 1) << 4, xn);
    compute_store(tile << 4, xb);
    if (wave == 0) __builtin_amdgcn_s_wait_tensorcnt(0);
    __syncthreads();
  }
#else
  // Fallback: synchronous LDS staging (matches the TDM LDS image).
  for (int tt = 0; tt < TPB; ++tt) {
    int fo0 = (tile0 + tt) << 4;
    __syncthreads();
    for (int tl = wave; tl < 34; tl += NWAVE) {
      int t = fo0 * 2 - 1 + tl;
      bool tok = (t >= 0) && (t < FIN);
      if constexpr (SRCMODE == 0) {
        const float* srow  = src  + ((size_t)b * FIN + t) * (NJ * 3);
        const float* orow  = src2 + ((size_t)b * FIN + t) * NJ;
        for (int q = lane; q < NJ * 3; q += 32)
          xlds0[tl * XIS + q] = tok ? srow[q] : 0.f;
        for (int q = lane; q < NJ; q += 32)
          xlds0[34 * XIS + tl * XOS + q] = tok ? orow[q] : 0.f;
      } else {
        const float* srow = src + ((size_t)b * FIN + t) * CIN;
        for (int q = lane; q < CIN; q += 32)
          xlds0[tl * XS + q] = tok ? srow[q] : 0.f;
      }
    }
    __syncthreads();
    compute_store(fo0, xlds0);
  }
#endif
}

// ---------------------------------------------------------------------------
// Graph block: one lane per (row, joint); row of 22 float4 staged in LDS.
// ---------------------------------------------------------------------------
__global__ __launch_bounds__(32 * GBY) void graph_kernel(
    const float* __restrict__ xin, float* __restrict__ xout, int nrows,
    const float* __restrict__ n2n_w, const float* __restrict__ n2n_b,
    const float* __restrict__ e2n_we, const float* __restrict__ e2n_wn,
    const float* __restrict__ e2n_b,  const float* __restrict__ n2e_wn,
    const float* __restrict__ n2e_we, const float* __restrict__ n2e_b,
    const float* __restrict__ lin_w,  const float* __restrict__ lin_b)
{
  __shared__ float4 rowbuf[GBY][NJ];
  const int lane = threadIdx.x;
  const int ry   = threadIdx.y;
  const long long row = (long long)blockIdx.x * GBY + ry;
  const bool rowok = row < nrows;

  if (rowok && lane < NJ)
    rowbuf[ry][lane] = ((const float4*)(xin + row * CIN))[lane];
  __syncthreads();
  if (!rowok || lane >= NJ) return;

  const int i = lane;
  constexpr int PAR[NJ] = {0,0,1,2,3,4,0,6,7,8,0,10,11,12,12,14,15,16,12,18,19,20};

  float an0 = 0.f, an1 = 0.f, an2 = 0.f, ae = 0.f;   // child-sum aggregation
  #pragma unroll
  for (int c = 1; c < NJ; ++c) {
    if (PAR[c] == i) {
      float4 vv = rowbuf[ry][c];
      an0 += vv.x; an1 += vv.y; an2 += vv.z; ae += vv.w;
    }
  }

  float4 me = rowbuf[ry][i];
  float pn0 = 0.f, pn1 = 0.f, pn2 = 0.f;             // node[parent(i)], 0 at root
  if (i > 0) {
    float4 pv = rowbuf[ry][d_PARENT[i]];
    pn0 = pv.x; pn1 = pv.y; pn2 = pv.z;
  }

  float f1[3], f2[3];
  #pragma unroll
  for (int d = 0; d < 3; ++d) {
    f1[d] = an0 * n2n_w[d] + an1 * n2n_w[3 + d] + an2 * n2n_w[6 + d] + n2n_b[d];
    f2[d] = ae * e2n_we[d]
          + me.x * e2n_wn[d] + me.y * e2n_wn[3 + d] + me.z * e2n_wn[6 + d] + e2n_b[d];
  }
  float ne = pn0 * n2e_wn[0] + pn1 * n2e_wn[1] + pn2 * n2e_wn[2]
           + me.w * n2e_we[0] + n2e_b[0];

  float h[3];
  #pragma unroll
  for (int d = 0; d < 3; ++d) {
    h[d] = f1[0] * lin_w[d]      + f1[1] * lin_w[3 + d]  + f1[2] * lin_w[6 + d]
         + f2[0] * lin_w[9 + d]  + f2[1] * lin_w[12 + d] + f2[2] * lin_w[15 + d]
         + lin_b[d];
  }

  float4 outv = make_float4(h[0], h[1], h[2], ne);
  ((float4*)(xout + row * CIN))[i] = outv;
}

// ---------------------------------------------------------------------------
extern "C" void kernel_launch(void* const* d_in, const int* in_sizes, int n_in,
                              void* d_out, int out_size, void* d_ws, size_t ws_size,
                              hipStream_t stream) {
  (void)in_sizes; (void)n_in; (void)out_size; (void)ws_size;
  const float* input  = (const float*)d_in[0];
  const float* offset = (const float*)d_in[1];
  const float* c1w = (const float*)d_in[2]; const float* c1b = (const float*)d_in[3];
  const float* c2w = (const float*)d_in[4]; const float* c2b = (const float*)d_in[5];
  const float* c3w = (const float*)d_in[6]; const float* c3b = (const float*)d_in[7];

  float* bufA = (float*)d_ws;                          // 128*1024*88 floats
  float* bufB = bufA + (size_t)128 * 1024 * CIN;       // ping-pong
  const size_t shmem =
      (size_t)(NROWS_W * WS + 2 * XBUFMAX + NROWS_W * YS) * sizeof(float);

  // conv1: F 2048 -> 1024, dual-TDM gather from (input, offset)
  conv_wmma_kernel<0, 0, 2048, 1024, 4>
      <<<dim3(128 * 16), dim3(BCONV), shmem, stream>>>(
          input, offset, c1w, c1b, bufA);
  // graph block 1 (weights d_in[8..17])
  {
    int nrows = 128 * 1024;
    graph_kernel<<<dim3(nrows / GBY), dim3(32, GBY), 0, stream>>>(
        bufA, bufB, nrows,
        (const float*)d_in[8],  (const float*)d_in[9],  (const float*)d_in[10],
        (const float*)d_in[11], (const float*)d_in[12], (const float*)d_in[13],
        (const float*)d_in[14], (const float*)d_in[15], (const float*)d_in[16],
        (const float*)d_in[17]);
  }
  // conv2: F 1024 -> 512
  conv_wmma_kernel<1, 0, 1024, 512, 4>
      <<<dim3(128 * 8), dim3(BCONV), shmem, stream>>>(
          bufB, nullptr, c2w, c2b, bufA);
  // graph block 2 (weights d_in[18..27])
  {
    int nrows = 128 * 512;
    graph_kernel<<<dim3(nrows / GBY), dim3(32, GBY), 0, stream>>>(
        bufA, bufB, nrows,
        (const float*)d_in[18], (const float*)d_in[19], (const float*)d_in[20],
        (const float*)d_in[21], (const float*)d_in[22], (const float*)d_in[23],
        (const float*)d_in[24], (const float*)d_in[25], (const float*)d_in[26],
        (const float*)d_in[27]);
  }
  // conv3: F 512 -> 256, write out[..., :3] directly
  conv_wmma_kernel<1, 1, 512, 256, 4>
      <<<dim3(128 * 4), dim3(BCONV), shmem, stream>>>(
          bufB, nullptr, c3w, c3b, (float*)d_out);
}